// BaseMPNNLayer_1_43044162240638
// MI455X (gfx1250) — compile-verified
//
#include <hip/hip_runtime.h>

// CDNA5 / gfx1250: wave32, WMMA 16x16 shapes.
typedef float v2f __attribute__((ext_vector_type(2)));
typedef float v8f __attribute__((ext_vector_type(8)));

#define FD 64   // input features D
#define FO 64   // output features DO

// ---------------------------------------------------------------------------
// out[v][f] = b[f]  (bias pre-folded so scatter can atomically accumulate),
// deg[v] = 0
// ---------------------------------------------------------------------------
__global__ void init_out_deg(float* __restrict__ out, float* __restrict__ deg,
                             const float* __restrict__ b, int N) {
  long long i = (long long)blockIdx.x * blockDim.x + threadIdx.x;
  long long total = (long long)N * FO;
  if (i < total) out[i] = b[(int)(i & (FO - 1))];
  if (i < N) deg[(int)i] = 0.0f;
}

// ---------------------------------------------------------------------------
// deg[dst] += 1   (in-degree; exact in f32 for counts < 2^24)
// ---------------------------------------------------------------------------
__global__ void degree_kernel(const int* __restrict__ E, float* __restrict__ deg,
                              int NE) {
  int e = blockIdx.x * blockDim.x + threadIdx.x;
  if (e < NE) {
    int t = E[NE + e];                 // E row 1 = dst
    unsafeAtomicAdd(&deg[t], 1.0f);    // native global_atomic_add_f32, no return
  }
}

// ---------------------------------------------------------------------------
// deg[v] = rsqrt(max(deg[v], 1))   (in place -> inv_sqrt_deg)
// ---------------------------------------------------------------------------
__global__ void rsqrt_kernel(float* __restrict__ deg, int N) {
  int i = blockIdx.x * blockDim.x + threadIdx.x;
  if (i < N) deg[i] = rsqrtf(fmaxf(deg[i], 1.0f));
}

// ---------------------------------------------------------------------------
// H = X @ W via V_WMMA_F32_16X16X4_F32 (fp32 WMMA: keeps reference precision;
// compute is ~1e-4 of the roofline so precision, not rate, picks the flavor).
// One wave handles one 16-row tile x all 64 output columns:
//   16 K-steps (K=4 each) x 4 column tiles = 64 WMMAs, acc in 4 x v8f.
// A layout (32-bit A 16x4): lanes 0-15 hold K=k0,k0+1; lanes 16-31 K=k0+2,k0+3.
// B layout (4x16): lane n<16 holds col n, K pair per VGPR; mirrored for hi half.
// C/D layout: VGPR r -> row r (lanes 0-15) / row r+8 (lanes 16-31).
// ---------------------------------------------------------------------------
__global__ void __launch_bounds__(256)
gemm_wmma(const float* __restrict__ X, const float* __restrict__ W,
          float* __restrict__ H, int N) {
  __shared__ float ldsW[FD * FO];      // 16 KB of 320 KB/WGP

  for (int i = threadIdx.x; i < FD * FO; i += blockDim.x)
    ldsW[i] = W[i];
  __syncthreads();

  const int wave = threadIdx.x >> 5;
  const int lane = threadIdx.x & 31;
  const int hi   = lane >> 4;          // 0: lanes 0-15, 1: lanes 16-31
  const int mrow = lane & 15;

  const int nTiles = (N + 15) >> 4;
  const int tile = blockIdx.x * (blockDim.x >> 5) + wave;
  if (tile >= nTiles) return;

  int row = tile * 16 + mrow;
  int rowc = row < N ? row : (N - 1);            // clamp tail loads
  const float* __restrict__ xr = X + (size_t)rowc * FD;

  v8f acc0 = {}, acc1 = {}, acc2 = {}, acc3 = {};

#pragma unroll 4
  for (int k0 = 0; k0 < FD; k0 += 4) {
    const int kk = k0 + hi * 2;                  // this half-wave's K pair
    v2f a;
    a.x = xr[kk];
    a.y = xr[kk + 1];

    const float* w0 = &ldsW[kk * FO + mrow];
    const float* w1 = &ldsW[(kk + 1) * FO + mrow];
    v2f b0, b1, b2, b3;
    b0.x = w0[0];  b0.y = w1[0];
    b1.x = w0[16]; b1.y = w1[16];
    b2.x = w0[32]; b2.y = w1[32];
    b3.x = w0[48]; b3.y = w1[48];

    acc0 = __builtin_amdgcn_wmma_f32_16x16x4_f32(false, a, false, b0, (short)0, acc0, false, false);
    acc1 = __builtin_amdgcn_wmma_f32_16x16x4_f32(false, a, false, b1, (short)0, acc1, false, false);
    acc2 = __builtin_amdgcn_wmma_f32_16x16x4_f32(false, a, false, b2, (short)0, acc2, false, false);
    acc3 = __builtin_amdgcn_wmma_f32_16x16x4_f32(false, a, false, b3, (short)0, acc3, false, false);
  }

#pragma unroll
  for (int r = 0; r < 8; ++r) {
    int grow = tile * 16 + r + hi * 8;           // C/D VGPR r -> row r (+8 hi)
    if (grow < N) {
      float* dst = H + (size_t)grow * FO + mrow;
      dst[0]  = acc0[r];
      dst[16] = acc1[r];
      dst[32] = acc2[r];
      dst[48] = acc3[r];
    }
  }
}

// ---------------------------------------------------------------------------
// Per-edge: out[dst] += H[src] * inv[src]*inv[dst].
// Half-wave (16 lanes) per edge; each lane moves 4 contiguous floats
// (global_load_b128 gather from L2-resident H, 4 no-return f32 atomics).
// ---------------------------------------------------------------------------
__global__ void edge_scatter(const int* __restrict__ E,
                             const float* __restrict__ H,
                             const float* __restrict__ inv,
                             float* __restrict__ out, int NE) {
  long long gid = (long long)blockIdx.x * blockDim.x + threadIdx.x;
  int e   = (int)(gid >> 4);           // 16 threads per edge
  int sub = threadIdx.x & 15;
  if (e >= NE) return;

  int s = E[e];                        // src
  int t = E[NE + e];                   // dst
  float nrm = inv[s] * inv[t];

  const float4* __restrict__ hs = (const float4*)(H + (size_t)s * FO);
  float4 h = hs[sub];

  float* o = out + (size_t)t * FO + sub * 4;
  unsafeAtomicAdd(o + 0, h.x * nrm);
  unsafeAtomicAdd(o + 1, h.y * nrm);
  unsafeAtomicAdd(o + 2, h.z * nrm);
  unsafeAtomicAdd(o + 3, h.w * nrm);
}

// ---------------------------------------------------------------------------
extern "C" void kernel_launch(void* const* d_in, const int* in_sizes, int n_in,
                              void* d_out, int out_size, void* d_ws, size_t ws_size,
                              hipStream_t stream) {
  // Inputs (setup_inputs order): V, E, X, W, b
  const int*   E = (const int*)d_in[1];    // (2, NE) int32 row-major
  const float* X = (const float*)d_in[2];  // (N, 64)
  const float* W = (const float*)d_in[3];  // (64, 64)
  const float* b = (const float*)d_in[4];  // (64,)
  float* out = (float*)d_out;              // (N, 64)

  const int N  = in_sizes[0];
  const int NE = in_sizes[1] / 2;

  // Workspace: [deg/inv_sqrt_deg : N floats][H : N*64 floats] (~26 MB; H fits L2)
  float* deg = (float*)d_ws;
  float* H   = deg + (((size_t)N + 1023) & ~(size_t)1023);

  // 1) out = bias, deg = 0
  {
    long long total = (long long)N * FO;
    int grid = (int)((total + 255) / 256);
    init_out_deg<<<grid, 256, 0, stream>>>(out, deg, b, N);
  }
  // 2) in-degree
  degree_kernel<<<(NE + 255) / 256, 256, 0, stream>>>(E, deg, NE);
  // 3) inv sqrt degree
  rsqrt_kernel<<<(N + 255) / 256, 256, 0, stream>>>(deg, N);
  // 4) H = X @ W  (fp32 WMMA; 8 waves/block, one 16x64 tile per wave)
  {
    int nTiles = (N + 15) / 16;
    int grid = (nTiles + 7) / 8;
    gemm_wmma<<<grid, 256, 0, stream>>>(X, W, H, N);
  }
  // 5) gather-scale-scatter over edges
  {
    long long threads = (long long)NE * 16;
    int grid = (int)((threads + 255) / 256);
    edge_scatter<<<grid, 256, 0, stream>>>(E, H, deg, out, NE);
  }
}